// NSbuilder_55963423867499
// MI455X (gfx1250) — compile-verified
//
#include <hip/hip_runtime.h>

// Problem constants (fixed by the reference: DEPTH=8, H=W=512)
#define LBITS 256
#define HH    512
#define WW    512
#define TB    32          // t-values per block (L / gridDim.y)
#define TPB   128         // threads per block: 128 * float4 = 512 = one row

typedef __attribute__((ext_vector_type(4))) float v4f;

__device__ __forceinline__ float ns_bit(float ft, float nsl, float ns, float st,
                                        float rt, const float* lds_rng) {
    const bool  m  = ft < nsl;                                   // non-stable phase?
    float d  = ft - nsl;                                         // integer-valued
    d        = fminf(fmaxf(d, 0.0f), 255.0f);                    // jnp.clip
    const float r  = m ? rt : lds_rng[(int)d];                   // rng draw
    const float th = m ? ns : st;                                // threshold
    return (th > r) ? 1.0f : 0.0f;
}

__global__ __launch_bounds__(TPB)
void nsbuilder_kernel(const float* __restrict__ src_ns,
                      const float* __restrict__ src_st,
                      const float* __restrict__ new_ns_len,
                      const float* __restrict__ rng,
                      float* __restrict__ out)
{
    __shared__ float lds_rng[LBITS];

    const int tid = threadIdx.x;        // 0..127

    // ---- CDNA5 async copy: rng[256] (1 KB) -> LDS, 8 bytes per lane ----
    // GVS mode: global addr = SADDR(64b SGPR pair) + VADDR(32b offset);
    // VDST VGPR holds the LDS byte address (low 32 bits of the flat shared ptr).
    {
        unsigned lds_off = (unsigned)(size_t)(&lds_rng[0]) + (unsigned)(tid * 8);
        unsigned g_off   = (unsigned)(tid * 8);
        asm volatile("global_load_async_to_lds_b64 %0, %1, %2"
                     :
                     : "v"(lds_off), "v"(g_off), "s"(rng)
                     : "memory");
        asm volatile("s_wait_asynccnt 0" ::: "memory");   // my wave's copy done
    }
    __syncthreads();                                      // all 4 waves' copies visible

    // ---- per-thread tile: one float4 of one row, TB consecutive t values ----
    const int    h     = blockIdx.x;          // 0..511
    const int    tBase = blockIdx.y * TB;     // 0,32,...,224
    const int    w4    = tid * 4;             // 0..508
    const size_t pix   = (size_t)h * WW + (size_t)w4;

    const v4f ns  = *(const v4f*)(src_ns     + pix);
    const v4f st  = *(const v4f*)(src_st     + pix);
    const v4f nsl = *(const v4f*)(new_ns_len + pix);

    float* __restrict__ outp = out + pix;

    #pragma unroll 4
    for (int tt = 0; tt < TB; ++tt) {
        const int   t  = tBase + tt;
        const float ft = (float)t;
        const float rt = lds_rng[t];          // same for all lanes (LDS broadcast)

        v4f o;
        o.x = ns_bit(ft, nsl.x, ns.x, st.x, rt, lds_rng);
        o.y = ns_bit(ft, nsl.y, ns.y, st.y, rt, lds_rng);
        o.z = ns_bit(ft, nsl.z, ns.z, st.z, rt, lds_rng);
        o.w = ns_bit(ft, nsl.w, ns.w, st.w, rt, lds_rng);

        // 256 MB write-once stream > 192 MB L2: non-temporal 128-bit store.
        __builtin_nontemporal_store(o, (v4f*)(outp + (size_t)t * (HH * WW)));
    }
}

extern "C" void kernel_launch(void* const* d_in, const int* in_sizes, int n_in,
                              void* d_out, int out_size, void* d_ws, size_t ws_size,
                              hipStream_t stream) {
    const float* src_ns     = (const float*)d_in[0];
    const float* src_st     = (const float*)d_in[1];
    const float* new_ns_len = (const float*)d_in[2];
    const float* rng        = (const float*)d_in[3];
    float* out = (float*)d_out;

    dim3 grid(HH, LBITS / TB);   // (512, 8) blocks
    nsbuilder_kernel<<<grid, TPB, 0, stream>>>(src_ns, src_st, new_ns_len, rng, out);
}